// GCNBlock_77300821393916
// MI455X (gfx1250) — compile-verified
//
#include <hip/hip_runtime.h>
#include <hip/hip_bf16.h>
#include <math.h>

// ---------------------------------------------------------------------------
// GATv2 x2 (+BN, tanh) for MI455X / gfx1250.
// Dense projections use V_WMMA_F32_16X16X4_F32 (fp32 matrix pipe, matches
// reference precision). Edge softmax/aggregation uses L2-resident gathers +
// hardware fp32 atomics (working set ~85MB << 192MB L2).
// ---------------------------------------------------------------------------

typedef __attribute__((ext_vector_type(2))) float v2f;
typedef __attribute__((ext_vector_type(8))) float v8f;

#define NEG_SLOPE 0.2f
#define BN_EPS 1e-5f

// ---------------------------------------------------------------------------
// FP32 WMMA GEMM: C[M x 128] = A[M x K] * B[K x 128]
// grid.x = M/16 row tiles; block = 256 threads = 8 waves; wave w owns column
// tile w (16 cols). A-frag layout (16x4 f32): lanes 0-15 hold (k,k+1) of row
// lane, lanes 16-31 hold (k+2,k+3). B-frag (4x16) mirrors with N on lanes.
// ---------------------------------------------------------------------------
__global__ void gemm_wmma_f32(const float* __restrict__ A,
                              const float* __restrict__ B,
                              float* __restrict__ C,
                              int M, int K) {
  const int Nn   = 128;
  const int wave = threadIdx.x >> 5;
  const int lane = threadIdx.x & 31;
  const int half = lane >> 4;       // 0: K pair (k,k+1) ; 1: (k+2,k+3)
  const int l    = lane & 15;       // row (A) / col (B) within tile
  const int m0   = blockIdx.x * 16;
  const int n0   = wave * 16;

  v8f c = {};
  for (int k = 0; k < K; k += 4) {
    const int ka = k + half * 2;
    v2f a, b;
    a.x = A[(size_t)(m0 + l) * K + ka];
    a.y = A[(size_t)(m0 + l) * K + ka + 1];
    b.x = B[(size_t)ka * Nn + n0 + l];
    b.y = B[(size_t)(ka + 1) * Nn + n0 + l];
    // (neg_a, A, neg_b, B, c_mod, C, reuse_a, reuse_b)
    c = __builtin_amdgcn_wmma_f32_16x16x4_f32(false, a, false, b,
                                              (short)0, c, false, false);
  }
  // D layout: VGPR j -> row m0+j (lanes 0-15) / m0+j+8 (lanes 16-31), col n0+l
#pragma unroll
  for (int j = 0; j < 8; ++j) {
    const int row = m0 + j + half * 8;
    C[(size_t)row * Nn + n0 + l] = c[j];
  }
}

// ---------------------------------------------------------------------------
// Helpers
// ---------------------------------------------------------------------------
__device__ __forceinline__ void atomicMaxF32(float* addr, float val) {
  // sign-separated monotonic int trick; addr initialized to -inf
  if (val >= 0.0f)
    atomicMax((int*)addr, __float_as_int(val));
  else
    atomicMin((unsigned int*)addr, __float_as_uint(val));
}

__device__ __forceinline__ void edge_endpoints(const int* __restrict__ ei,
                                               int eid, int E,
                                               int& src, int& dst) {
  if (eid < E) { src = ei[eid]; dst = ei[E + eid]; }
  else         { src = dst = eid - E; }         // self loops appended
}

__global__ void fill_f32(float* p, float v, int n) {
  int t = blockIdx.x * blockDim.x + threadIdx.x;
  if (t < n) p[t] = v;
}

// ---------------------------------------------------------------------------
// Pass A: e[eid][h] = sum_c leakyrelu(xp[src]+xp[dst]) * a ; atomicMax emax
// One wave per edge; lane -> (head = lane>>4, 4 channels = (lane&15)*4).
// ---------------------------------------------------------------------------
__global__ void edge_score(const float* __restrict__ xp,
                           const int* __restrict__ ei,
                           const float* __restrict__ avec,
                           float* __restrict__ e,
                           float* __restrict__ emax,
                           int E, int Etot) {
  const int eid = blockIdx.x * (blockDim.x >> 5) + (threadIdx.x >> 5);
  if (eid >= Etot) return;
  const int lane = threadIdx.x & 31;
  int src, dst; edge_endpoints(ei, eid, E, src, dst);

  const int h   = lane >> 4;
  const int idx = (lane & 15) * 4;
  const float4 s  = *(const float4*)(xp + (size_t)src * 128 + h * 64 + idx);
  const float4 d  = *(const float4*)(xp + (size_t)dst * 128 + h * 64 + idx);
  const float4 av = *(const float4*)(avec + h * 64 + idx);

  float m, lr, acc = 0.0f;
  m = s.x + d.x; lr = (m > 0.0f) ? m : NEG_SLOPE * m; acc += lr * av.x;
  m = s.y + d.y; lr = (m > 0.0f) ? m : NEG_SLOPE * m; acc += lr * av.y;
  m = s.z + d.z; lr = (m > 0.0f) ? m : NEG_SLOPE * m; acc += lr * av.z;
  m = s.w + d.w; lr = (m > 0.0f) ? m : NEG_SLOPE * m; acc += lr * av.w;

  // reduce across the 16 lanes of this head
  for (int off = 8; off > 0; off >>= 1) acc += __shfl_down(acc, off, 16);
  if ((lane & 15) == 0) {
    e[(size_t)eid * 2 + h] = acc;
    atomicMaxF32(&emax[(size_t)dst * 2 + h], acc);
  }
}

// ---------------------------------------------------------------------------
// Pass B: ee = exp(e - emax[dst]); denom[dst] += ee    (thread per edge,head)
// ---------------------------------------------------------------------------
__global__ void edge_exp(const int* __restrict__ ei,
                         float* __restrict__ e,
                         const float* __restrict__ emax,
                         float* __restrict__ denom,
                         int E, int Etot) {
  int t = blockIdx.x * blockDim.x + threadIdx.x;
  if (t >= Etot * 2) return;
  const int eid = t >> 1, h = t & 1;
  int src, dst; edge_endpoints(ei, eid, E, src, dst);
  (void)src;
  const float ee = expf(e[t] - emax[(size_t)dst * 2 + h]);
  e[t] = ee;
  __hip_atomic_fetch_add(&denom[(size_t)dst * 2 + h], ee,
                         __ATOMIC_RELAXED, __HIP_MEMORY_SCOPE_AGENT);
}

// ---------------------------------------------------------------------------
// Pass C: acc[dst] += alpha * xp[src]   (wave per edge, f32 hardware atomics)
// ---------------------------------------------------------------------------
__global__ void edge_agg(const float* __restrict__ xp,
                         const int* __restrict__ ei,
                         const float* __restrict__ e,
                         const float* __restrict__ denom,
                         float* __restrict__ acc,
                         int E, int Etot) {
  const int eid = blockIdx.x * (blockDim.x >> 5) + (threadIdx.x >> 5);
  if (eid >= Etot) return;
  const int lane = threadIdx.x & 31;
  int src, dst; edge_endpoints(ei, eid, E, src, dst);

  const int h   = lane >> 4;
  const int idx = (lane & 15) * 4;
  const float alpha = e[(size_t)eid * 2 + h] /
                      (denom[(size_t)dst * 2 + h] + 1e-16f);
  const float4 s = *(const float4*)(xp + (size_t)src * 128 + h * 64 + idx);
  float* p = acc + (size_t)dst * 128 + h * 64 + idx;
  __hip_atomic_fetch_add(p + 0, alpha * s.x, __ATOMIC_RELAXED, __HIP_MEMORY_SCOPE_AGENT);
  __hip_atomic_fetch_add(p + 1, alpha * s.y, __ATOMIC_RELAXED, __HIP_MEMORY_SCOPE_AGENT);
  __hip_atomic_fetch_add(p + 2, alpha * s.z, __ATOMIC_RELAXED, __HIP_MEMORY_SCOPE_AGENT);
  __hip_atomic_fetch_add(p + 3, alpha * s.w, __ATOMIC_RELAXED, __HIP_MEMORY_SCOPE_AGENT);
}

// ---------------------------------------------------------------------------
// Head mean + BN statistics (LDS-reduced partial sums, few global atomics).
// tmp[n][c] = 0.5*(acc[n][c] + acc[n][64+c]);  sum/sumsq per channel.
// ---------------------------------------------------------------------------
__global__ void head_mean_stats(const float* __restrict__ acc,
                                float* __restrict__ tmp,
                                float* __restrict__ sum,
                                float* __restrict__ sumsq,
                                int N) {
  __shared__ float s1[256];
  __shared__ float s2[256];
  const int c = threadIdx.x & 63;
  const int nodesPerBlk = blockDim.x >> 6;         // 4
  float ls = 0.0f, lq = 0.0f;
  for (int n = blockIdx.x * nodesPerBlk + (threadIdx.x >> 6);
       n < N; n += gridDim.x * nodesPerBlk) {
    const float v = 0.5f * (acc[(size_t)n * 128 + c] +
                            acc[(size_t)n * 128 + 64 + c]);
    tmp[(size_t)n * 64 + c] = v;
    ls += v; lq += v * v;
  }
  s1[threadIdx.x] = ls; s2[threadIdx.x] = lq;
  __syncthreads();
  if (threadIdx.x < 64) {
    ls = s1[threadIdx.x] + s1[threadIdx.x + 64] +
         s1[threadIdx.x + 128] + s1[threadIdx.x + 192];
    lq = s2[threadIdx.x] + s2[threadIdx.x + 64] +
         s2[threadIdx.x + 128] + s2[threadIdx.x + 192];
    __hip_atomic_fetch_add(&sum[c],   ls, __ATOMIC_RELAXED, __HIP_MEMORY_SCOPE_AGENT);
    __hip_atomic_fetch_add(&sumsq[c], lq, __ATOMIC_RELAXED, __HIP_MEMORY_SCOPE_AGENT);
  }
}

__global__ void bn_apply(const float* __restrict__ tmp,
                         const float* __restrict__ sum,
                         const float* __restrict__ sumsq,
                         const float* __restrict__ gamma,
                         const float* __restrict__ beta,
                         float* __restrict__ out,
                         int N, int doTanh) {
  int t = blockIdx.x * blockDim.x + threadIdx.x;
  if (t >= N * 64) return;
  const int c = t & 63;
  const float invN = 1.0f / (float)N;
  const float mean = sum[c] * invN;
  const float var  = sumsq[c] * invN - mean * mean;
  float v = (tmp[t] - mean) * rsqrtf(var + BN_EPS) * gamma[c] + beta[c];
  out[t] = doTanh ? tanhf(v) : v;
}

// ---------------------------------------------------------------------------
// Host-side orchestration
// ---------------------------------------------------------------------------
extern "C" void kernel_launch(void* const* d_in, const int* in_sizes, int n_in,
                              void* d_out, int out_size, void* d_ws, size_t ws_size,
                              hipStream_t stream) {
  const float* x      = (const float*)d_in[0];
  const int*   ei     = (const int*)  d_in[1];
  const float* W0     = (const float*)d_in[2];
  const float* a0     = (const float*)d_in[3];
  const float* gamma0 = (const float*)d_in[4];
  const float* beta0  = (const float*)d_in[5];
  const float* W1     = (const float*)d_in[6];
  const float* a1     = (const float*)d_in[7];
  const float* gamma1 = (const float*)d_in[8];
  const float* beta1  = (const float*)d_in[9];
  float* out = (float*)d_out;

  const int N    = in_sizes[0] / 128;   // 50000
  const int E    = in_sizes[1] / 2;     // 800000
  const int Etot = E + N;               // with self loops

  // workspace carve-up (floats), 256B aligned
  char* ws = (char*)d_ws;
  size_t off = 0;
  auto carve = [&](size_t nfloats) {
    float* p = (float*)(ws + off);
    off = (off + nfloats * sizeof(float) + 255) & ~(size_t)255;
    return p;
  };
  float* xp    = carve((size_t)N * 128);   // projected features [N,128]
  float* acc   = carve((size_t)N * 128);   // aggregation accumulator
  float* e     = carve((size_t)Etot * 2);  // edge scores / exp scores
  float* emax  = carve((size_t)N * 2);
  float* denom = carve((size_t)N * 2);
  float* tmp   = carve((size_t)N * 64);    // head-mean, pre-BN
  float* h0    = carve((size_t)N * 64);    // layer-0 output
  float* stats = carve(128);               // sum[64] ++ sumsq[64]
  (void)ws_size;

  const int waveEdgeBlocks = (Etot + 7) / 8;        // 8 edges per 256-thr block
  const int ehBlocks       = (Etot * 2 + 255) / 256;
  const int nodeElBlocks   = (N * 64 + 255) / 256;
  const int fillBlocks     = (N * 2 + 255) / 256;

  // ---------------- Layer 0: GATv2 -> BN -> tanh ----------------
  gemm_wmma_f32<<<N / 16, 256, 0, stream>>>(x, W0, xp, N, 128);

  fill_f32<<<fillBlocks, 256, 0, stream>>>(emax, -INFINITY, N * 2);
  hipMemsetAsync(denom, 0, (size_t)N * 2 * sizeof(float), stream);
  hipMemsetAsync(acc,   0, (size_t)N * 128 * sizeof(float), stream);
  hipMemsetAsync(stats, 0, 128 * sizeof(float), stream);

  edge_score<<<waveEdgeBlocks, 256, 0, stream>>>(xp, ei, a0, e, emax, E, Etot);
  edge_exp  <<<ehBlocks,       256, 0, stream>>>(ei, e, emax, denom, E, Etot);
  edge_agg  <<<waveEdgeBlocks, 256, 0, stream>>>(xp, ei, e, denom, acc, E, Etot);

  head_mean_stats<<<256, 256, 0, stream>>>(acc, tmp, stats, stats + 64, N);
  bn_apply<<<nodeElBlocks, 256, 0, stream>>>(tmp, stats, stats + 64,
                                             gamma0, beta0, h0, N, 1);

  // ---------------- Layer 1: GATv2 -> BN ----------------
  gemm_wmma_f32<<<N / 16, 256, 0, stream>>>(h0, W1, xp, N, 64);

  fill_f32<<<fillBlocks, 256, 0, stream>>>(emax, -INFINITY, N * 2);
  hipMemsetAsync(denom, 0, (size_t)N * 2 * sizeof(float), stream);
  hipMemsetAsync(acc,   0, (size_t)N * 128 * sizeof(float), stream);
  hipMemsetAsync(stats, 0, 128 * sizeof(float), stream);

  edge_score<<<waveEdgeBlocks, 256, 0, stream>>>(xp, ei, a1, e, emax, E, Etot);
  edge_exp  <<<ehBlocks,       256, 0, stream>>>(ei, e, emax, denom, E, Etot);
  edge_agg  <<<waveEdgeBlocks, 256, 0, stream>>>(xp, ei, e, denom, acc, E, Etot);

  head_mean_stats<<<256, 256, 0, stream>>>(acc, tmp, stats, stats + 64, N);
  bn_apply<<<nodeElBlocks, 256, 0, stream>>>(tmp, stats, stats + 64,
                                             gamma1, beta1, out, N, 0);
}